// CPPRefine_24232205484094
// MI455X (gfx1250) — compile-verified
//
#include <hip/hip_runtime.h>
#include <hip/hip_bf16.h>
#include <math.h>

typedef __attribute__((ext_vector_type(2))) float v2f;
typedef __attribute__((ext_vector_type(8))) float v8f;

#define BATCH 4
#define RAYS 32
#define CHAN 256
#define HDIM 256
#define WDIM 256
#define HW   (HDIM * WDIM)
#define KERO 6

// ---------------------------------------------------------------------------
// Kernel 1: conf0[b,r,hw] = sum_c w0[r,c] * feat[b,c,hw]
// One wave per 16-wide pixel tile; M=32 via two accumulators sharing one B tile.
// V_WMMA_F32_16X16X4_F32, K-loop of 64 steps.
// Layouts (ISA 7.12.2):
//   A 16x4 f32 : VGPR0 = K0(l<16)/K2(l>=16), VGPR1 = K1/K3, row = lane%16
//   B 4x16 f32 : symmetric, col = lane%16, lane-half selects K pair
//   D 16x16    : VGPR v -> row v + 8*(lane>=16), col = lane%16
// ---------------------------------------------------------------------------
__global__ __launch_bounds__(256) void conf0_gemm_kernel(
    const float* __restrict__ feat,   // (B, C, H, W)
    const float* __restrict__ w0,     // (R, C)
    float* __restrict__ conf0)        // (B, R, H, W)
{
    const int gwave = (blockIdx.x * blockDim.x + threadIdx.x) >> 5;
    const int lane  = threadIdx.x & 31;
    const int b     = gwave >> 12;            // HW/16 = 4096 tiles per batch
    const int tile  = gwave & 4095;
    const int hw0   = tile << 4;
    const int hi    = lane >> 4;              // which K half this lane holds
    const int ln    = lane & 15;

    const float* fb = feat + (size_t)b * CHAN * HW;

    v8f acc_lo = {};
    v8f acc_hi = {};

    for (int c = 0; c < CHAN; c += 4) {
        const int cr = c + 2 * hi;            // this lane's K rows: cr, cr+1
        // A tiles from w0 (rows r, contiguous K pair -> 64-bit loads, L2-resident)
        v2f a_lo = *(const v2f*)(w0 + (size_t)ln * CHAN + cr);
        v2f a_hi = *(const v2f*)(w0 + (size_t)(ln + 16) * CHAN + cr);
        // B tile from features (K rows strided by HW)
        v2f bm;
        bm.x = fb[(size_t)cr * HW + hw0 + ln];
        bm.y = fb[(size_t)(cr + 1) * HW + hw0 + ln];

        acc_lo = __builtin_amdgcn_wmma_f32_16x16x4_f32(
            false, a_lo, false, bm, (short)0, acc_lo, false, false);
        acc_hi = __builtin_amdgcn_wmma_f32_16x16x4_f32(
            false, a_hi, false, bm, (short)0, acc_hi, false, false);
    }

    float* cb = conf0 + (size_t)b * RAYS * HW;
#pragma unroll
    for (int v = 0; v < 8; ++v) {
        const int row = v + 8 * hi;
        cb[(size_t)row * HW + hw0 + ln]        = acc_lo[v];
        cb[(size_t)(row + 16) * HW + hw0 + ln] = acc_hi[v];
    }
}

// ---------------------------------------------------------------------------
// Kernel 2: per-pixel refine. One thread per (b, r, h, w).
// ---------------------------------------------------------------------------
__global__ __launch_bounds__(256) void refine_kernel(
    const float* __restrict__ sd,     // (B, R, H, W)
    const float* __restrict__ conf0,  // (B, R, H, W)
    const float* __restrict__ w1,     // (K, K) row-major [j][k]
    const float* __restrict__ b1,     // (K,)
    float* __restrict__ ray_out,      // (B, R, H, W)
    float* __restrict__ conf_out)     // (B, K, R, H, W)
{
    const size_t idx = (size_t)blockIdx.x * blockDim.x + threadIdx.x;
    const int hw = (int)(idx & (HW - 1));
    const int br = (int)(idx >> 16);          // HW = 2^16
    const int r  = br & (RAYS - 1);
    const int b  = br >> 5;
    const int wx = hw & (WDIM - 1);
    const int hy = hw >> 8;                   // WDIM = 2^8

    // angle = (r / R) * 2*pi, as in reference (f32 throughout)
    const float ang = ((float)r * 0.03125f) * 6.28318530717958647692f;
    const float ca = cosf(ang);
    const float sa = sinf(ang);

    const float* psd = sd    + (size_t)br * HW;
    const float* pc0 = conf0 + (size_t)br * HW;

    const float s  = psd[hw];
    const float c0 = pc0[hw];

    const float ef[5] = {0.2f, 0.4f, 0.6f, 0.8f, 1.0f};
    float rays[KERO];
    float confs[KERO];
    rays[0]  = s;
    confs[0] = c0;

    const float fx = (float)wx;
    const float fy = (float)hy;

#pragma unroll
    for (int k = 0; k < 5; ++k) {
        const float base = (s - 1.0f) * ef[k];
        // reference index math, verbatim sequence
        const float gx = fx + ca * base;
        const float gy = fy + sa * base;
        const float nx = gx / (float)(WDIM - 1) * 2.0f - 1.0f;
        const float ny = gy / (float)(HDIM - 1) * 2.0f - 1.0f;
        const float ix = ((nx + 1.0f) * (float)WDIM - 1.0f) * 0.5f;
        const float iy = ((ny + 1.0f) * (float)HDIM - 1.0f) * 0.5f;
        const int ixr = (int)rintf(ix);       // round half to even -> v_rndne
        const int iyr = (int)rintf(iy);
        const bool valid = (ixr >= 0) & (ixr < WDIM) & (iyr >= 0) & (iyr < HDIM);
        int ixc = ixr < 0 ? 0 : (ixr > WDIM - 1 ? WDIM - 1 : ixr);
        int iyc = iyr < 0 ? 0 : (iyr > HDIM - 1 ? HDIM - 1 : iyr);
        const int g = iyc * WDIM + ixc;
        const float ssd = valid ? psd[g] : 0.0f;
        const float sc0 = valid ? pc0[g] : 0.0f;
        rays[k + 1]  = ssd + base;
        confs[k + 1] = sc0;
    }

    // conv_1_confidence: logit[j] = b1[j] + sum_k w1[j,k] * confs[k]
    float logit[KERO];
#pragma unroll
    for (int j = 0; j < KERO; ++j) {
        float acc = b1[j];
#pragma unroll
        for (int k = 0; k < KERO; ++k)
            acc = fmaf(w1[j * KERO + k], confs[k], acc);
        logit[j] = acc;
    }

    // softmax over j
    float m = logit[0];
#pragma unroll
    for (int j = 1; j < KERO; ++j) m = fmaxf(m, logit[j]);
    float e[KERO];
    float sum = 0.0f;
#pragma unroll
    for (int j = 0; j < KERO; ++j) { e[j] = expf(logit[j] - m); sum += e[j]; }
    const float inv = 1.0f / sum;

    float ray = 0.0f;
#pragma unroll
    for (int j = 0; j < KERO; ++j) {
        const float p = e[j] * inv;
        ray = fmaf(rays[j], p, ray);
        conf_out[(((size_t)b * KERO + j) * RAYS + r) * HW + hw] = p;
    }
    ray_out[(size_t)br * HW + hw] = fmaxf(ray, 0.0f);
}

// ---------------------------------------------------------------------------
extern "C" void kernel_launch(void* const* d_in, const int* in_sizes, int n_in,
                              void* d_out, int out_size, void* d_ws, size_t ws_size,
                              hipStream_t stream) {
    const float* sd   = (const float*)d_in[0];   // (B,R,H,W)
    const float* feat = (const float*)d_in[1];   // (B,C,H,W)
    const float* w0   = (const float*)d_in[2];   // (R,C)
    const float* w1   = (const float*)d_in[3];   // (K,K)
    const float* b1   = (const float*)d_in[4];   // (K,)

    float* out   = (float*)d_out;
    float* conf0 = (float*)d_ws;                 // B*R*HW floats = 33.6 MB scratch

    // Kernel 1: one wave (32 thr) per 16-pixel tile; B * HW/16 = 16384 waves
    const int waves   = BATCH * (HW / 16);
    const int threads = waves * 32;
    conf0_gemm_kernel<<<dim3(threads / 256), dim3(256), 0, stream>>>(feat, w0, conf0);

    // Kernel 2: one thread per (b, r, pixel)
    const size_t total = (size_t)BATCH * RAYS * HW;   // 8,388,608
    refine_kernel<<<dim3((unsigned)(total / 256)), dim3(256), 0, stream>>>(
        sd, conf0, w1, b1, out, out + total);
}